// DiagonalSISOCell_47708496724570
// MI455X (gfx1250) — compile-verified
//
#include <hip/hip_runtime.h>
#include <hip/hip_bf16.h>

typedef float v2f __attribute__((ext_vector_type(2)));
typedef float v8f __attribute__((ext_vector_type(8)));

#define D_DIM 64
#define N_DIM 16

// ---------------- Kernel 1: delta[v] = softplus(x[v,:] . w + b) ----------------
__global__ __launch_bounds__(256)
void delta_kernel(const float* __restrict__ x, const float* __restrict__ w,
                  const float* __restrict__ bias, float* __restrict__ delta, int V) {
    int v = blockIdx.x * blockDim.x + threadIdx.x;
    if (v >= V) return;
    const float4* xr = (const float4*)(x + (size_t)v * D_DIM);
    const float4* wr = (const float4*)w;
    float acc = 0.f;
#pragma unroll
    for (int i = 0; i < D_DIM / 4; ++i) {
        float4 xv = xr[i];
        float4 wv = wr[i];
        acc += xv.x * wv.x + xv.y * wv.y + xv.z * wv.z + xv.w * wv.w;
    }
    float z = acc + bias[0];
    // numerically-stable softplus
    float sp = (z > 20.f) ? z : __logf(1.f + __expf(z));
    delta[v] = sp;
}

// ---------------- Kernel 2: ZOH update + per-d 16x16 contraction via WMMA ----------------
// Workgroup: 256 threads (8 wave32). Tile = 16 consecutive v's; wave w owns d in [8w, 8w+8).
// Per (tile, d): D(16x16) = NS(16x16) @ C[d](16x16) using 4x V_WMMA_F32_16X16X4_F32 (K=4 each).
__global__ __launch_bounds__(256)
void ssm_wmma_kernel(const float* __restrict__ x, const float* __restrict__ state,
                     const float* __restrict__ log_nA, const float* __restrict__ Bm,
                     const float* __restrict__ Cm, const float* __restrict__ delta,
                     float* __restrict__ out) {
    __shared__ float A_lds[D_DIM * N_DIM]; // -exp(log_nA)
    __shared__ float B_lds[D_DIM * N_DIM];

    int t = threadIdx.x;
    // Stage A = -exp(log_nA) and B into LDS: 1024 elems, 4 per thread.
#pragma unroll
    for (int i = 0; i < 4; ++i) {
        int idx = t + 256 * i;
        A_lds[idx] = -__expf(log_nA[idx]);
        B_lds[idx] = Bm[idx];
    }
    __syncthreads();

    const int lane = t & 31;
    const int wave = t >> 5;    // 0..7
    const int row  = lane & 15; // M index (v within tile) for A; q column for B/D
    const int hi   = lane >> 4; // selects K sub-pair per the 16x4 f32 A layout

    const long long v0 = (long long)blockIdx.x * 16;
    const long long v  = v0 + row;
    const float dl = delta[v];

    for (int dd = 0; dd < 8; ++dd) {
        const int d = wave * 8 + dd;
        const float* srow = state + (v * D_DIM + d) * N_DIM;
        const float  xv   = x[v * D_DIM + d];
        const float* crow = Cm + d * (N_DIM * N_DIM);

        v2f a[4];
        v2f b[4];
#pragma unroll
        for (int c = 0; c < 4; ++c) {
            // A-matrix 16x4 f32 layout: lanes 0-15 hold K={k0,k0+1}, lanes 16-31 K={k0+2,k0+3}
            const int n = 4 * c + 2 * hi;
            float2 s2 = *(const float2*)(srow + n);
            const float an0 = A_lds[d * N_DIM + n];
            const float an1 = A_lds[d * N_DIM + n + 1];
            const float bn0 = B_lds[d * N_DIM + n];
            const float bn1 = B_lds[d * N_DIM + n + 1];
            // new_state = exp(delta*A)*state + delta*B*x
            a[c].x = __expf(dl * an0) * s2.x + dl * bn0 * xv;
            a[c].y = __expf(dl * an1) * s2.y + dl * bn1 * xv;
            // B-matrix 4x16 (K x N): row K=n of C[d], column = lane's q. Rows of C[d]
            // are contiguous over q -> 64B-coalesced per lane half.
            b[c].x = crow[n * N_DIM + row];
            b[c].y = crow[(n + 1) * N_DIM + row];
        }

        v8f acc = {};
        // (neg_a, A, neg_b, B, c_mod, C, reuse_a, reuse_b)
        acc = __builtin_amdgcn_wmma_f32_16x16x4_f32(false, a[0], false, b[0], (short)0, acc, false, false);
        acc = __builtin_amdgcn_wmma_f32_16x16x4_f32(false, a[1], false, b[1], (short)0, acc, false, false);
        acc = __builtin_amdgcn_wmma_f32_16x16x4_f32(false, a[2], false, b[2], (short)0, acc, false, false);
        acc = __builtin_amdgcn_wmma_f32_16x16x4_f32(false, a[3], false, b[3], (short)0, acc, false, false);

        // C/D layout: VGPR j -> M = j + 8*hi (lanes 16-31 hold M=8..15), N = row
#pragma unroll
        for (int j = 0; j < 8; ++j) {
            const int m = j + 8 * hi;
            out[((v0 + m) * D_DIM + d) * N_DIM + row] = acc[j];
        }
    }
}

extern "C" void kernel_launch(void* const* d_in, const int* in_sizes, int n_in,
                              void* d_out, int out_size, void* d_ws, size_t ws_size,
                              hipStream_t stream) {
    const float* x       = (const float*)d_in[0]; // [V, 64]
    const float* state   = (const float*)d_in[1]; // [V, 64, 16]
    const float* log_nA  = (const float*)d_in[2]; // [64, 16]
    const float* B       = (const float*)d_in[3]; // [64, 16]
    const float* C       = (const float*)d_in[4]; // [64, 16, 16]
    const float* w_delta = (const float*)d_in[5]; // [1, 64]
    const float* b_delta = (const float*)d_in[6]; // [1]
    float* out   = (float*)d_out;
    float* delta = (float*)d_ws; // V floats of scratch

    const int V = in_sizes[0] / D_DIM;

    delta_kernel<<<(V + 255) / 256, 256, 0, stream>>>(x, w_delta, b_delta, delta, V);
    ssm_wmma_kernel<<<V / 16, 256, 0, stream>>>(x, state, log_nA, B, C, delta, out);
}